// GCNBlock_9698036155164
// MI455X (gfx1250) — compile-verified
//
#include <hip/hip_runtime.h>
#include <hip/hip_bf16.h>

typedef __attribute__((ext_vector_type(2))) float v2f;
typedef __attribute__((ext_vector_type(8))) float v8f;

#define FDIM 128
#define LDA  132   // padded LDS row stride (132 mod 64 = 4 -> conflict-free frag reads)

// ---------------- degree / dinv ----------------
__global__ __launch_bounds__(256) void deg_init_kernel(float* deg, int n) {
    int i = blockIdx.x * 256 + threadIdx.x;
    if (i < n) deg[i] = 1.0f;                      // self loop
}

__global__ __launch_bounds__(256) void deg_accum_kernel(const int* __restrict__ dst,
                                                        float* deg, int ne) {
    int e = blockIdx.x * 256 + threadIdx.x;
    if (e < ne) atomicAdd(&deg[dst[e]], 1.0f);
}

__global__ __launch_bounds__(256) void deg_finalize_kernel(float* dinv, int n) {
    int i = blockIdx.x * 256 + threadIdx.x;
    if (i < n) dinv[i] = __frsqrt_rn(dinv[i]);     // deg >= 1
}

// ---------------- fp32 WMMA GEMM: H[M,128] = X[M,128] @ W[128,128] ----------------
// Persistent waves: block = 256 threads (8 waves). Wave w owns N-tile w and keeps
// its entire B slice (128x16 f32 = 32 x v2f frags) in VGPRs for its whole lifetime.
// Inner loop per 16-row block: async A staging (memory -> LDS, ASYNCcnt) + 32 WMMAs.
__global__ __launch_bounds__(256)
void gemm_f32_wmma_kernel(const float* __restrict__ X, const float* __restrict__ W,
                          float* __restrict__ H, int nblocks) {
    __shared__ float As[16 * LDA];

    const int wave = threadIdx.x >> 5;     // 0..7 => N tile
    const int lane = threadIdx.x & 31;
    const int half = lane >> 4;            // 0: K pair {0,1}, 1: K pair {2,3}
    const int l15  = lane & 15;            // A row / B col within tile

    // ---- load B fragments into registers once (W is 64KB, L2-resident) ----
    v2f breg[32];
    {
        const float* Wcol = W + wave * 16 + l15 + (size_t)(2 * half) * FDIM;
#pragma unroll
        for (int kk = 0; kk < 32; ++kk) {
            breg[kk].x = Wcol[(size_t)(4 * kk) * FDIM];
            breg[kk].y = Wcol[(size_t)(4 * kk + 1) * FDIM];
        }
    }

    // per-thread staging addresses: 256 threads x 32B (two async b128 ops)
    const int tr = threadIdx.x >> 4;              // 16 rows
    const int tc = (threadIdx.x & 15) * 8;        // 8 floats per thread
    unsigned ldsoff = (unsigned)(uintptr_t)(As + tr * LDA + tc);
    const float* Arow = As + l15 * LDA + 2 * half;

    for (int blk = blockIdx.x; blk < nblocks; blk += gridDim.x) {
        __syncthreads();   // all waves done reading As from previous iteration

        // ---- async global -> LDS copy of the 16x128 A block ----
        unsigned long long gaddr =
            (unsigned long long)(uintptr_t)(X + (size_t)blk * 16 * FDIM + tr * FDIM + tc);
        asm volatile("global_load_async_to_lds_b128 %0, %1, off"
                     :: "v"(ldsoff), "v"(gaddr) : "memory");
        asm volatile("global_load_async_to_lds_b128 %0, %1, off offset:16"
                     :: "v"(ldsoff), "v"(gaddr) : "memory");
        asm volatile("s_wait_asynccnt 0" ::: "memory");
        __syncthreads();   // every wave's async data visible in LDS

        // ---- 16x16x128 via 32 chained V_WMMA_F32_16X16X4_F32 ----
        v8f c = {0.f, 0.f, 0.f, 0.f, 0.f, 0.f, 0.f, 0.f};
#pragma unroll
        for (int kk = 0; kk < 32; ++kk) {
            v2f a;
            a.x = Arow[4 * kk];
            a.y = Arow[4 * kk + 1];
            c = __builtin_amdgcn_wmma_f32_16x16x4_f32(false, a, false, breg[kk],
                                                      (short)0, c, false, false);
        }

        // C/D layout: VGPR v -> M = v + 8*half, N = l15
        float* Hblk = H + (size_t)blk * 16 * FDIM + wave * 16;
        const int rb = half * 8;
#pragma unroll
        for (int v = 0; v < 8; ++v)
            Hblk[(size_t)(rb + v) * FDIM + l15] = c[v];
    }
}

// ---------------- aggregation ----------------
// out = h * dinv^2 (self loop) + bias   (also serves as the zero-init of out)
__global__ __launch_bounds__(256)
void agg_init_kernel(const float* __restrict__ H, const float* __restrict__ dinv,
                     const float* __restrict__ bias, float* __restrict__ out, int n) {
    int idx = blockIdx.x * 256 + threadIdx.x;
    if (idx >= n * FDIM) return;
    int row = idx >> 7, col = idx & (FDIM - 1);
    float di = dinv[row];
    out[idx] = H[idx] * di * di + bias[col];
}

// one wave32 per edge: lane moves 4 floats; 4x global_atomic_add_f32 per lane.
// H and out are L2-resident (20MB each in 192MB L2), so this runs at L2 rate.
__global__ __launch_bounds__(256)
void agg_edge_kernel(const float* __restrict__ H, const int* __restrict__ src,
                     const int* __restrict__ dst, const float* __restrict__ dinv,
                     float* __restrict__ out, int ne) {
    int e = (blockIdx.x * 256 + threadIdx.x) >> 5;
    int lane = threadIdx.x & 31;
    if (e >= ne) return;
    int s = src[e];
    int d = dst[e];
    float norm = dinv[s] * dinv[d];
    const float4 v = *(const float4*)(H + (size_t)s * FDIM + lane * 4);
    float* o = out + (size_t)d * FDIM + lane * 4;
    atomicAdd(o + 0, v.x * norm);
    atomicAdd(o + 1, v.y * norm);
    atomicAdd(o + 2, v.z * norm);
    atomicAdd(o + 3, v.w * norm);
}

// ---------------- batch norm (training stats) + relu ----------------
// 128 threads/block: thread = one column; coalesced row-major reads.
__global__ __launch_bounds__(128)
void bn_stats_kernel(const float* __restrict__ x, float* __restrict__ stats, int n) {
    int col = threadIdx.x;
    float s = 0.f, sq = 0.f;
    for (int row = blockIdx.x; row < n; row += gridDim.x) {
        float v = x[(size_t)row * FDIM + col];
        s += v;
        sq += v * v;
    }
    atomicAdd(&stats[col], s);
    atomicAdd(&stats[FDIM + col], sq);
}

__global__ __launch_bounds__(256)
void bn_relu_kernel(float* __restrict__ x, const float* __restrict__ stats,
                    const float* __restrict__ gamma, const float* __restrict__ beta,
                    int n) {
    int idx = blockIdx.x * 256 + threadIdx.x;
    if (idx >= n * FDIM) return;
    int col = idx & (FDIM - 1);
    float inv_n = 1.0f / (float)n;
    float mean = stats[col] * inv_n;
    float var  = stats[FDIM + col] * inv_n - mean * mean;
    float y = gamma[col] * (x[idx] - mean) * __frsqrt_rn(var + 1e-5f) + beta[col];
    x[idx] = fmaxf(y, 0.0f);
}

// ---------------- driver ----------------
extern "C" void kernel_launch(void* const* d_in, const int* in_sizes, int n_in,
                              void* d_out, int out_size, void* d_ws, size_t ws_size,
                              hipStream_t stream) {
    const float* x   = (const float*)d_in[0];
    const int*   ei  = (const int*)d_in[1];
    const float* W1  = (const float*)d_in[2];
    const float* b1  = (const float*)d_in[3];
    const float* g1  = (const float*)d_in[4];
    const float* be1 = (const float*)d_in[5];
    const float* W2  = (const float*)d_in[6];
    const float* b2  = (const float*)d_in[7];
    const float* g2  = (const float*)d_in[8];
    const float* be2 = (const float*)d_in[9];

    const int N = in_sizes[0] / FDIM;      // 40000
    const int E = in_sizes[1] / 2;         // 640000
    const int* src = ei;
    const int* dst = ei + E;

    float* out   = (float*)d_out;
    float* ws    = (float*)d_ws;
    float* dinv  = ws;                     // [N]
    float* stats = ws + N;                 // [256]
    float* tmp   = ws + N + 256;           // [N*128]

    const int b256   = 256;
    const int gN     = (N + 255) / 256;
    const int gE     = (E + 255) / 256;
    const int gNF    = (N * FDIM + 255) / 256;
    const int gEdgeW = (E * 32 + 255) / 256;   // one wave per edge
    const int nblk   = N / 16;                 // 2500 row blocks
    const int gGemm  = (nblk < 640) ? nblk : 640;  // persistent grid

    // degrees (shared by both layers)
    deg_init_kernel<<<gN, b256, 0, stream>>>(dinv, N);
    deg_accum_kernel<<<gE, b256, 0, stream>>>(dst, dinv, E);
    deg_finalize_kernel<<<gN, b256, 0, stream>>>(dinv, N);

    // ---------- layer 1 ----------
    gemm_f32_wmma_kernel<<<gGemm, b256, 0, stream>>>(x, W1, tmp, nblk);
    agg_init_kernel<<<gNF, b256, 0, stream>>>(tmp, dinv, b1, out, N);
    agg_edge_kernel<<<gEdgeW, b256, 0, stream>>>(tmp, src, dst, dinv, out, E);
    hipMemsetAsync(stats, 0, 2 * FDIM * sizeof(float), stream);
    bn_stats_kernel<<<512, FDIM, 0, stream>>>(out, stats, N);
    bn_relu_kernel<<<gNF, b256, 0, stream>>>(out, stats, g1, be1, N);

    // ---------- layer 2 (gemm consumes `out` before agg_init overwrites it) ----------
    gemm_f32_wmma_kernel<<<gGemm, b256, 0, stream>>>(out, W2, tmp, nblk);
    agg_init_kernel<<<gNF, b256, 0, stream>>>(tmp, dinv, b2, out, N);
    agg_edge_kernel<<<gEdgeW, b256, 0, stream>>>(tmp, src, dst, dinv, out, E);
    hipMemsetAsync(stats, 0, 2 * FDIM * sizeof(float), stream);
    bn_stats_kernel<<<512, FDIM, 0, stream>>>(out, stats, N);
    bn_relu_kernel<<<gNF, b256, 0, stream>>>(out, stats, g2, be2, N);
}